// MultiHeadDiffAttention_50208167690401
// MI455X (gfx1250) — compile-verified
//
#include <hip/hip_runtime.h>
#include <hip/hip_bf16.h>
#include <math.h>

typedef float v2f __attribute__((ext_vector_type(2)));
typedef float v8f __attribute__((ext_vector_type(8)));

#define B_SZ   2
#define T_SEQ  2048
#define C_DIM  1024
#define NH     16      // value heads
#define HD     32      // q/k head dim
#define QKH    32      // 2*NH q/k sub-heads
#define VD     64      // value head dim
#define ROWQKV 3072

__device__ __forceinline__ v8f wmma4(v2f a, v2f b, v8f c) {
  // V_WMMA_F32_16X16X4_F32: D(16x16 f32) = A(16x4 f32) * B(4x16 f32) + C
  return __builtin_amdgcn_wmma_f32_16x16x4_f32(
      /*neg_a=*/false, a, /*neg_b=*/false, b,
      /*c_mod=*/(short)0, c, /*reuse_a=*/false, /*reuse_b=*/false);
}

// ---------------------------------------------------------------------------
// fp32 WMMA GEMM: C[M,N] = A[M,K] * B[K,N], all row-major.
// Block = 128x128 tile, 256 threads = 8 waves (4 row-groups x 2 col-groups),
// each wave a 32x64 register tile: 2 A-frags x 4 B-frags -> 8 chained WMMAs
// per 4-deep K chunk. Software-pipelined: next chunk's fragments are loaded
// before the current chunk's WMMA burst; addresses advance by increment only.
// Requires M%128==0, N%128==0, K%8==0.
// ---------------------------------------------------------------------------
__global__ __launch_bounds__(256) void k_gemm(const float* __restrict__ A,
                                              const float* __restrict__ Bm,
                                              float* __restrict__ Cm,
                                              int N, int K) {
  const int lane = threadIdx.x & 31;
  const int wid  = threadIdx.x >> 5;
  const int wy   = wid & 3;              // 4 row groups of 32
  const int wx   = wid >> 2;             // 2 col groups of 64
  const int rowBase = blockIdx.y * 128 + wy * 32;
  const int colBase = blockIdx.x * 128 + wx * 64;
  const int hi   = lane >> 4;
  const int l15  = lane & 15;
  const int bc   = colBase + l15;

  const float* pA0 = A + (size_t)(rowBase + l15) * K + hi * 2;       // rows 0..15
  const float* pA1 = A + (size_t)(rowBase + 16 + l15) * K + hi * 2;  // rows 16..31
  const float* pB  = Bm + (size_t)(hi * 2) * N + bc;
  const size_t strideB = (size_t)4 * N;

  v8f acc[2][4] = {};

  // prologue: fragments for chunk 0
  v2f a0 = *(const v2f*)pA0;
  v2f a1 = *(const v2f*)pA1;
  v2f b[4];
  #pragma unroll
  for (int cg = 0; cg < 4; ++cg) {
    b[cg].x = pB[16 * cg];
    b[cg].y = pB[N + 16 * cg];
  }

  #pragma unroll 2
  for (int k = 0; k < K - 4; k += 4) {
    pA0 += 4; pA1 += 4; pB += strideB;
    // issue next chunk's loads first so they overlap the WMMA burst
    const v2f na0 = *(const v2f*)pA0;
    const v2f na1 = *(const v2f*)pA1;
    v2f nb[4];
    #pragma unroll
    for (int cg = 0; cg < 4; ++cg) {
      nb[cg].x = pB[16 * cg];
      nb[cg].y = pB[N + 16 * cg];
    }
    // stream B ~16 chunks ahead into WGP$/L2 (speculative; tail drop is safe)
    if ((k & 60) == 0) {
      __builtin_prefetch(pB + 16 * strideB, 0, 1);
      __builtin_prefetch(pA0 + 64, 0, 1);
      __builtin_prefetch(pA1 + 64, 0, 1);
    }
    #pragma unroll
    for (int cg = 0; cg < 4; ++cg) {
      acc[0][cg] = wmma4(a0, b[cg], acc[0][cg]);
      acc[1][cg] = wmma4(a1, b[cg], acc[1][cg]);
    }
    a0 = na0; a1 = na1;
    #pragma unroll
    for (int cg = 0; cg < 4; ++cg) b[cg] = nb[cg];
  }
  // epilogue chunk
  #pragma unroll
  for (int cg = 0; cg < 4; ++cg) {
    acc[0][cg] = wmma4(a0, b[cg], acc[0][cg]);
    acc[1][cg] = wmma4(a1, b[cg], acc[1][cg]);
  }

  // C layout per 16x16 tile: VGPR r holds row M = r + hi*8, col N = l15
  #pragma unroll
  for (int g = 0; g < 2; ++g) {
    #pragma unroll
    for (int r = 0; r < 8; ++r) {
      const size_t row = (size_t)(rowBase + g * 16 + r + hi * 8);
      #pragma unroll
      for (int cg = 0; cg < 4; ++cg) {
        Cm[row * N + bc + 16 * cg] = acc[g][cg][r];
      }
    }
  }
}

// ---------------------------------------------------------------------------
// In-place RoPE over the q and k regions of qkv (B,T,3072).
// One thread per (b, t, qk, head, pair-j); pairs are (2j, 2j+1) within dim 32.
// ---------------------------------------------------------------------------
__global__ void k_rope(float* __restrict__ qkv) {
  size_t idx = (size_t)blockIdx.x * blockDim.x + threadIdx.x;
  const int j  = idx & 15;  idx >>= 4;          // 16 pairs
  const int h  = idx & 31;  idx >>= 5;          // 32 sub-heads
  const int qk = idx & 1;   idx >>= 1;          // 0 = q, 1 = k
  const int t  = idx & 2047; idx >>= 11;        // 2048 timesteps
  const int b  = (int)idx;                      // 2 batches
  const float theta = __expf(-((float)(2 * j) / 32.0f) * logf(10000.0f));
  float s, c;
  __sincosf((float)t * theta, &s, &c);
  float* p = qkv + ((size_t)(b * T_SEQ + t)) * ROWQKV + qk * C_DIM + h * HD + 2 * j;
  const float x1 = p[0], x2 = p[1];
  p[0] = x1 * c - x2 * s;
  p[1] = x1 * s + x2 * c;
}

// ---------------------------------------------------------------------------
// lambda = exp(dot(lq1,lk1)) - exp(dot(lq2,lk2)) + 0.2   (vectors of length 32)
// ---------------------------------------------------------------------------
__global__ void k_lambda(const float* __restrict__ lq1, const float* __restrict__ lk1,
                         const float* __restrict__ lq2, const float* __restrict__ lk2,
                         float* __restrict__ lam) {
  const int l = threadIdx.x;
  float s1 = lq1[l] * lk1[l];
  float s2 = lq2[l] * lk2[l];
  #pragma unroll
  for (int off = 16; off; off >>= 1) {
    s1 += __shfl_xor(s1, off, 32);
    s2 += __shfl_xor(s2, off, 32);
  }
  if (l == 0) *lam = __expf(s1) - __expf(s2) + 0.2f;
}

// ---------------------------------------------------------------------------
// Flash attention, one wave per (query-tile of 16, sub-head h in [0,32), b).
// QK^T via 8 chained f32 WMMAs, online softmax through LDS, P*V via 16 WMMAs.
// Output A[b][h][t][0..63] to Abuf.
// ---------------------------------------------------------------------------
__global__ __launch_bounds__(32) void k_attn(const float* __restrict__ qkv,
                                             float* __restrict__ Abuf) {
  const int qt   = blockIdx.x;        // 128 query tiles
  const int h    = blockIdx.y;        // 32 sub-heads
  const int b    = blockIdx.z;        // 2 batches
  const int lane = threadIdx.x;
  const int hi   = lane >> 4;
  const int l15  = lane & 15;

  const float* Q  = qkv + (size_t)b * T_SEQ * ROWQKV + h * HD;
  const float* Kp = qkv + (size_t)b * T_SEQ * ROWQKV + C_DIM + h * HD;
  const float* V  = qkv + (size_t)b * T_SEQ * ROWQKV + 2 * C_DIM + (h & 15) * VD;

  __shared__ float S[16 * 18];        // stride 18: bank-conflict-free, 8B aligned
  __shared__ float mrow[16], lrow[16], frow[16];

  if (lane < 16) { mrow[lane] = -3.0e38f; lrow[lane] = 0.0f; }
  __syncthreads();

  // Q fragments for all 8 K-chunks of head dim 32 (A-operand layout)
  const size_t qoff = (size_t)(qt * 16 + l15) * ROWQKV;
  v2f qa[8];
  #pragma unroll
  for (int c = 0; c < 8; ++c) {
    const int col = 4 * c + hi * 2;
    qa[c].x = Q[qoff + col];
    qa[c].y = Q[qoff + col + 1];
  }

  v8f o0 = {}, o1 = {}, o2 = {}, o3 = {};
  const float scale = 0.17677669529663687f;   // 1/sqrt(32)

  for (int kb = 0; kb <= qt; ++kb) {
    // ---- scores S = Q * K^T (16x16) ----
    v8f s = {};
    const size_t koff = (size_t)(kb * 16 + l15) * ROWQKV;
    #pragma unroll
    for (int c = 0; c < 8; ++c) {
      const int col = 4 * c + hi * 2;
      v2f kf; kf.x = Kp[koff + col]; kf.y = Kp[koff + col + 1];
      s = wmma4(qa[c], kf, s);
    }
    // scale + causal mask, spill to LDS in row-major [m][n]
    #pragma unroll
    for (int r = 0; r < 8; ++r) {
      const int m    = r + hi * 8;
      const int qidx = qt * 16 + m;
      const int kidx = kb * 16 + l15;
      S[m * 18 + l15] = (kidx <= qidx) ? s[r] * scale : -3.0e38f;
    }
    __syncthreads();

    // ---- online softmax: lanes 0..15 each own one query row ----
    if (lane < 16) {
      float mx = mrow[lane];
      #pragma unroll
      for (int j = 0; j < 16; ++j) mx = fmaxf(mx, S[lane * 18 + j]);
      const float f = __expf(mrow[lane] - mx);
      float sum = 0.0f;
      #pragma unroll
      for (int j = 0; j < 16; ++j) {
        const float p = __expf(S[lane * 18 + j] - mx);
        S[lane * 18 + j] = p;
        sum += p;
      }
      mrow[lane] = mx;
      frow[lane] = f;
      lrow[lane] = lrow[lane] * f + sum;
    }
    __syncthreads();

    // ---- rescale running output ----
    float fr[8];
    #pragma unroll
    for (int r = 0; r < 8; ++r) fr[r] = frow[r + hi * 8];
    #pragma unroll
    for (int r = 0; r < 8; ++r) {
      o0[r] *= fr[r]; o1[r] *= fr[r]; o2[r] *= fr[r]; o3[r] *= fr[r];
    }

    // ---- P fragments (A-operand layout) from LDS ----
    v2f pa[4];
    #pragma unroll
    for (int c = 0; c < 4; ++c) {
      const int col = 4 * c + hi * 2;
      pa[c].x = S[l15 * 18 + col];
      pa[c].y = S[l15 * 18 + col + 1];
    }
    // ---- O += P * V  (V block 16 keys x 64 dims, 4 N-tiles) ----
    #pragma unroll
    for (int c = 0; c < 4; ++c) {
      const int krow = kb * 16 + 4 * c + hi * 2;
      const float* Vr0 = V + (size_t)krow * ROWQKV;
      const float* Vr1 = V + (size_t)(krow + 1) * ROWQKV;
      v2f vb0; vb0.x = Vr0[l15];      vb0.y = Vr1[l15];
      v2f vb1; vb1.x = Vr0[l15 + 16]; vb1.y = Vr1[l15 + 16];
      v2f vb2; vb2.x = Vr0[l15 + 32]; vb2.y = Vr1[l15 + 32];
      v2f vb3; vb3.x = Vr0[l15 + 48]; vb3.y = Vr1[l15 + 48];
      o0 = wmma4(pa[c], vb0, o0);
      o1 = wmma4(pa[c], vb1, o1);
      o2 = wmma4(pa[c], vb2, o2);
      o3 = wmma4(pa[c], vb3, o3);
    }
    __syncthreads();   // S reused next iteration
  }

  // ---- normalize by row sums and store A tile ----
  float inv[8];
  #pragma unroll
  for (int r = 0; r < 8; ++r) inv[r] = 1.0f / lrow[r + hi * 8];
  float* Ab = Abuf + ((size_t)(b * QKH + h)) * T_SEQ * VD;
  #pragma unroll
  for (int r = 0; r < 8; ++r) {
    float* Ar = Ab + (size_t)(qt * 16 + r + hi * 8) * VD;
    Ar[l15]      = o0[r] * inv[r];
    Ar[l15 + 16] = o1[r] * inv[r];
    Ar[l15 + 32] = o2[r] * inv[r];
    Ar[l15 + 48] = o3[r] * inv[r];
  }
}

// ---------------------------------------------------------------------------
// diff = A1 - lam*A2 ; RMS-norm over 64 ; *0.8 ; write (B,T,C) for projection.
// One wave per row of 64 (2 elements/lane), 8 rows per 256-thread block.
// ---------------------------------------------------------------------------
__global__ __launch_bounds__(256) void k_diffnorm(const float* __restrict__ Abuf,
                                                  const float* __restrict__ lamp,
                                                  float* __restrict__ Dbuf) {
  const float lam = *lamp;
  const int wid  = threadIdx.x >> 5;
  const int lane = threadIdx.x & 31;
  const size_t row = (size_t)blockIdx.x * 8 + wid;   // over B*NH*T = 65536
  const int t  = (int)(row & 2047);
  const size_t bh = row >> 11;
  const int hh = (int)(bh & 15);
  const int b  = (int)(bh >> 4);

  const float* A1 = Abuf + ((size_t)(b * QKH + hh))      * T_SEQ * VD + (size_t)t * VD;
  const float* A2 = Abuf + ((size_t)(b * QKH + 16 + hh)) * T_SEQ * VD + (size_t)t * VD;
  const float d0 = A1[lane]      - lam * A2[lane];
  const float d1 = A1[lane + 32] - lam * A2[lane + 32];
  float ss = d0 * d0 + d1 * d1;
  #pragma unroll
  for (int off = 16; off; off >>= 1) ss += __shfl_xor(ss, off, 32);
  const float r = rsqrtf(ss * (1.0f / 64.0f) + 1e-5f) * 0.8f;
  float* Dp = Dbuf + ((size_t)(b * T_SEQ + t)) * C_DIM + hh * VD;
  Dp[lane]      = d0 * r;
  Dp[lane + 32] = d1 * r;
}

// ---------------------------------------------------------------------------
extern "C" void kernel_launch(void* const* d_in, const int* in_sizes, int n_in,
                              void* d_out, int out_size, void* d_ws, size_t ws_size,
                              hipStream_t stream) {
  const float* x      = (const float*)d_in[0];
  const float* W_attn = (const float*)d_in[1];
  const float* W_proj = (const float*)d_in[2];
  const float* lq1    = (const float*)d_in[3];
  const float* lk1    = (const float*)d_in[4];
  const float* lq2    = (const float*)d_in[5];
  const float* lk2    = (const float*)d_in[6];
  float* out = (float*)d_out;

  float* qkv  = (float*)d_ws;                               // 2*2048*3072 floats
  float* Abuf = qkv  + (size_t)B_SZ * T_SEQ * ROWQKV;       // 2*32*2048*64 floats
  float* Dbuf = Abuf + (size_t)B_SZ * QKH * T_SEQ * VD;     // 2*2048*1024 floats
  float* lam  = Dbuf + (size_t)B_SZ * T_SEQ * C_DIM;        // 1 float

  const int M = B_SZ * T_SEQ;                               // 4096

  // 1. qkv = x @ W_attn  (4096 x 3072 x 1024)
  {
    dim3 grid(3 * C_DIM / 128, M / 128);
    k_gemm<<<grid, 256, 0, stream>>>(x, W_attn, qkv, 3 * C_DIM, C_DIM);
  }
  // 2. RoPE on q,k in place
  {
    const size_t n = (size_t)B_SZ * T_SEQ * 2 * QKH * 16;   // 4,194,304 threads
    k_rope<<<(unsigned)(n / 256), 256, 0, stream>>>(qkv);
  }
  // 3. lambda scalar
  k_lambda<<<1, 32, 0, stream>>>(lq1, lk1, lq2, lk2, lam);
  // 4. flash attention
  {
    dim3 grid(T_SEQ / 16, QKH, B_SZ);
    k_attn<<<grid, 32, 0, stream>>>(qkv, Abuf);
  }
  // 5. differential combine + RMS norm
  k_diffnorm<<<(B_SZ * NH * T_SEQ) / 8, 256, 0, stream>>>(Abuf, lam, Dbuf);
  // 6. out = Dbuf @ W_proj  (4096 x 1024 x 1024)
  {
    dim3 grid(C_DIM / 128, M / 128);
    k_gemm<<<grid, 256, 0, stream>>>(Dbuf, W_proj, out, C_DIM, C_DIM);
  }
  (void)in_sizes; (void)n_in; (void)out_size; (void)ws_size;
}